// STAR_47605417509306
// MI455X (gfx1250) — compile-verified
//
#include <hip/hip_runtime.h>

typedef __attribute__((ext_vector_type(2))) float v2f;
typedef __attribute__((ext_vector_type(8))) float v8f;

#define ALPHA_C 0.7f
#define QDEC_C  0.3f

#define B_DIM   32
#define L_DIM   4096
#define C_DIM   512
#define CHUNK   512
#define NCHUNK  (L_DIM / CHUNK)        // 8
#define WARMUP  32
#define WPB     8                      // waves per block
#define CTILES  (C_DIM / 16)           // 32

__device__ __forceinline__ float qpow_i(int e) {
    float r = 1.0f;
    for (int i = 0; i < e; ++i) r *= QDEC_C;
    return r;
}

// EMA-scan via WMMA: each wave owns a 16-channel strip of one (batch, L-chunk).
// Within the chunk, 16-timestep blocks are computed as a lower-triangular
// 16x16 matmul (4 chained V_WMMA_F32_16X16X4_F32, K=16) with the carry
// injected into the accumulator and extracted from row 15 via shfl.
__global__ __launch_bounds__(256) void STAR_ema_wmma_kernel(
    const float* __restrict__ x, const float* __restrict__ gate_p,
    float* __restrict__ out)
{
    const int lane = threadIdx.x & 31;
    const int wave = threadIdx.x >> 5;
    const int half = lane >> 4;          // lane half (0: lanes 0-15, 1: 16-31)
    const int n    = lane & 15;          // A: row M (t_out). B/D: column N (channel)

    int bid = blockIdx.x;
    const int cgroup = bid % (CTILES / WPB);   // 4
    bid /= (CTILES / WPB);
    const int ci = bid % NCHUNK;
    const int b  = bid / NCHUNK;

    const int ctile = cgroup * WPB + wave;
    const int c = ctile * 16 + n;

    const float g = fminf(fmaxf(gate_p[0], 0.0f), 1.0f);
    const float one_m_2g = 1.0f - 2.0f * g;

    const float* __restrict__ xb = x   + (size_t)b * L_DIM * C_DIM;  // [L][C]
    float*       __restrict__ ob = out + (size_t)b * L_DIM * C_DIM;

    // --- A matrix (constant weights W[n][k] = (k<=n) ? alpha*q^(n-k) : 0)
    // A 16x4 f32 layout: VGPR0 = K (half?2:0), VGPR1 = K (half?3:1), M = lane&15.
    v2f wa[4];
    #pragma unroll
    for (int j = 0; j < 4; ++j) {
        const int k0 = 4 * j + 2 * half;
        wa[j][0] = (k0     <= n) ? ALPHA_C * qpow_i(n - k0)     : 0.0f;
        wa[j][1] = (k0 + 1 <= n) ? ALPHA_C * qpow_i(n - k0 - 1) : 0.0f;
    }

    // carry-injection factors: D VGPR r holds row M = r + 8*half -> q^(M+1)
    float cf[8];
    {
        float f = qpow_i(1 + 8 * half);
        #pragma unroll
        for (int r = 0; r < 8; ++r) { cf[r] = f; f *= QDEC_C; }
    }

    // --- warm-up: sequential EMA to converge carry ~= trend[t0-1]
    // (error <= q^(WARMUP-1) ~ 6e-17, far below fp32 eps). For chunk 0,
    // carry := x[0] makes trend[0] = a*x0 + q*x0 = x0 (a+q == 1).
    const int t0 = ci * CHUNK;
    const int ts = (ci == 0) ? 0 : (t0 - WARMUP);
    float carry = xb[(size_t)ts * C_DIM + c];
    for (int t = ts + 1; t < t0; ++t)
        carry = fmaf(QDEC_C, carry, ALPHA_C * xb[(size_t)t * C_DIM + c]);

    // --- main loop: 16-timestep blocks
    for (int tb = t0; tb < t0 + CHUNK; tb += 16) {
        const float* __restrict__ xt = xb + (size_t)tb * C_DIM;

        // B slabs (X): K x N layout -> VGPR0: K = 4j + 2*half, VGPR1: K+1
        v2f xbv[4];
        #pragma unroll
        for (int j = 0; j < 4; ++j) {
            const int k0 = 4 * j + 2 * half;
            xbv[j][0] = xt[(size_t)k0       * C_DIM + c];
            xbv[j][1] = xt[(size_t)(k0 + 1) * C_DIM + c];
        }
        // X re-read in D layout for the blend (WGP$ hits, no extra HBM traffic)
        float xd[8];
        #pragma unroll
        for (int r = 0; r < 8; ++r)
            xd[r] = xt[(size_t)(r + 8 * half) * C_DIM + c];

        // stream-ahead prefetch of next block (global_prefetch_b8)
        if (tb + 16 < t0 + CHUNK)
            __builtin_prefetch(xt + (size_t)16 * C_DIM + c, 0, 1);

        // accumulator init with carry term: d[r] = q^(M+1) * carry
        v8f d;
        #pragma unroll
        for (int r = 0; r < 8; ++r) d[r] = cf[r] * carry;

        // trend = W * X + carry-term : 4 chained fp32 WMMAs (K = 16)
        #pragma unroll
        for (int j = 0; j < 4; ++j)
            d = __builtin_amdgcn_wmma_f32_16x16x4_f32(
                    false, wa[j], false, xbv[j], (short)0, d, false, false);

        // next carry = trend[tb+15][c] : D VGPR7, lanes 16..31
        carry = __shfl(d[7], 16 + n, 32);

        // blend + coalesced store: out = g*x + (1-2g)*trend
        #pragma unroll
        for (int r = 0; r < 8; ++r) {
            const float o = fmaf(one_m_2g, d[r], g * xd[r]);
            ob[(size_t)(tb + r + 8 * half) * C_DIM + c] = o;
        }
    }
}

extern "C" void kernel_launch(void* const* d_in, const int* in_sizes, int n_in,
                              void* d_out, int out_size, void* d_ws, size_t ws_size,
                              hipStream_t stream) {
    const float* x    = (const float*)d_in[0];
    const float* gate = (const float*)d_in[1];
    float* out = (float*)d_out;
    (void)in_sizes; (void)n_in; (void)out_size; (void)d_ws; (void)ws_size;

    const int grid = B_DIM * NCHUNK * (CTILES / WPB);   // 32*8*4 = 1024 blocks
    STAR_ema_wmma_kernel<<<grid, WPB * 32, 0, stream>>>(x, gate, out);
}